// GLMSelfAttention_73409581023716
// MI455X (gfx1250) — compile-verified
//
#include <hip/hip_runtime.h>
#include <hip/hip_bf16.h>

#define BB 64
#define HH 32
#define HKVN 2
#define DD 128
#define HIDN 4096
#define QKVN (HH * DD + 2 * HKVN * DD)   // 4608
#define NBLKTAB 256                      // MAX_S / BLOCK
#define SPLITS 16
#define TILES_PER_SPLIT (NBLKTAB / SPLITS)  // 16 tiles of 16 keys
#define PART_STRIDE (16 * 128 + 32)      // acc + m[16] + l[16]

typedef __attribute__((ext_vector_type(2))) float v2f;
typedef __attribute__((ext_vector_type(8))) float v8f;

__device__ __forceinline__ v8f wmma_f32(v2f a, v2f b, v8f c) {
    // V_WMMA_F32_16X16X4_F32 : D = A(16x4) * B(4x16) + C(16x16)
    return __builtin_amdgcn_wmma_f32_16x16x4_f32(
        /*neg_a=*/false, a, /*neg_b=*/false, b,
        /*c_mod=*/(short)0, c, /*reuse_a=*/false, /*reuse_b=*/false);
}

// ---------------------------------------------------------------------------
// GEMM: C[M,N] = A[M,K] * W[N,K]^T   (one wave per 16x16 tile)
// ---------------------------------------------------------------------------
__global__ void __launch_bounds__(32)
gemm_wmma_kernel(const float* __restrict__ A, const float* __restrict__ W,
                 float* __restrict__ C, int Ndim, int Kdim) {
    const int mt = blockIdx.x, nt = blockIdx.y;
    const int lane = threadIdx.x;
    const int half = lane >> 4;
    const int lid  = lane & 15;
    const float* arow = A + (size_t)(mt * 16 + lid) * Kdim;   // A row m=lid
    const float* brow = W + (size_t)(nt * 16 + lid) * Kdim;   // B col n=lid (weight row)
    const int koff = half * 2;
    v8f acc = {};
    for (int k = 0; k < Kdim; k += 4) {
        v2f a = *(const v2f*)(arow + k + koff);
        v2f b = *(const v2f*)(brow + k + koff);
        acc = wmma_f32(a, b, acc);
    }
    // C layout: comp i -> row mt*16 + i + half*8, col nt*16 + lid
    float* ccol = C + (size_t)(nt * 16 + lid);
    const int mbase = mt * 16 + half * 8;
#pragma unroll
    for (int i = 0; i < 8; ++i)
        ccol[(size_t)(mbase + i) * Ndim] = acc[i];
}

// ---------------------------------------------------------------------------
// RoPE on q (in place) and k (in place + export), export v
// ---------------------------------------------------------------------------
__global__ void __launch_bounds__(1024)
rope_kernel(float* __restrict__ qkv, float* __restrict__ knew,
            float* __restrict__ vnew, const int* __restrict__ lens) {
    const int b = blockIdx.x;
    const int t = threadIdx.x;               // 0..1023
    const float pos = (float)(lens[b] - 1);
    float* base = qkv + (size_t)b * QKVN;

    // q: 32 heads x 32 interleaved pairs (first ROT=64 dims of each head)
    {
        const int head = t >> 5, p = t & 31;
        const float ang = pos * __powf(10000.0f, -(float)p / 32.0f);
        const float c = __cosf(ang), s = __sinf(ang);
        float* x = base + head * DD + 2 * p;
        const float x0 = x[0], x1 = x[1];
        x[0] = x0 * c - x1 * s;
        x[1] = x1 * c + x0 * s;
    }
    // k: 2 kv heads x 32 pairs
    if (t < HKVN * 32) {
        const int head = t >> 5, p = t & 31;
        const float ang = pos * __powf(10000.0f, -(float)p / 32.0f);
        const float c = __cosf(ang), s = __sinf(ang);
        float* x = base + HH * DD + head * DD + 2 * p;
        const float x0 = x[0], x1 = x[1];
        x[0] = x0 * c - x1 * s;
        x[1] = x1 * c + x0 * s;
    }
    __syncthreads();
    if (t < HKVN * DD) {
        knew[(size_t)b * HKVN * DD + t] = base[HH * DD + t];
        vnew[(size_t)b * HKVN * DD + t] = base[HH * DD + HKVN * DD + t];
    }
}

// ---------------------------------------------------------------------------
// Flash-decode paged attention: one wave per (batch, kv head, split)
// scores = Q(16x128) x K^T via 32 WMMAs / tile ; PV via 32 WMMAs / tile
// ---------------------------------------------------------------------------
__global__ void __launch_bounds__(32)
attn_kernel(const float* __restrict__ qkv, const float* __restrict__ knew,
            const float* __restrict__ vnew, const float* __restrict__ kc,
            const float* __restrict__ vc, const int* __restrict__ bt,
            const int* __restrict__ lens, float* __restrict__ part) {
    const int b = blockIdx.x, kv = blockIdx.y, sp = blockIdx.z;
    const int lane = threadIdx.x;
    const int half = lane >> 4, lid = lane & 15;
    const int len = lens[b];
    const int last = len - 1;
    const float scale = 0.08838834764831845f;  // 128^-0.5
    const float NEGINF = -__builtin_inff();
    __shared__ float plds[16 * 16];

    float* pb = part + ((size_t)(b * HKVN + kv) * SPLITS + sp) * PART_STRIDE;

    // Q A-fragments: row g = lid, dims chunked by 4 (upper half-wave: +2)
    const float* qrow = qkv + (size_t)b * QKVN + (size_t)(kv * 16 + lid) * DD;
    v2f aq[32];
#pragma unroll
    for (int c = 0; c < 32; ++c)
        aq[c] = *(const v2f*)(qrow + 4 * c + 2 * half);

    v8f acc[8];
#pragma unroll
    for (int i = 0; i < 8; ++i) acc[i] = (v8f){};
    float m[8], l[8];
#pragma unroll
    for (int i = 0; i < 8; ++i) { m[i] = NEGINF; l[i] = 0.f; }

    const float* knrow = knew + ((size_t)b * HKVN + kv) * DD;
    const float* vnrow = vnew + ((size_t)b * HKVN + kv) * DD;

    const int t0 = sp * TILES_PER_SPLIT;
    const int t1 = t0 + TILES_PER_SPLIT;
    for (int t = t0; t < t1; ++t) {
        const int s0 = t * 16;
        if (s0 >= len) break;
        const size_t flatbase = (size_t)bt[b * NBLKTAB + t] * 16;

        // ---- scores: 32 WMMAs over the 128 head dims ----
        const int skey = s0 + lid;  // this lane's key column
        const float* krow = (skey == last)
            ? knrow : kc + ((flatbase + lid) * HKVN + kv) * (size_t)DD;
        v8f sc = {};
#pragma unroll
        for (int c = 0; c < 32; ++c) {
            v2f bf = *(const v2f*)(krow + 4 * c + 2 * half);
            sc = wmma_f32(aq[c], bf, sc);
        }
        const bool valid = skey < len;

        // ---- online softmax (row stats across 16-lane half-wave groups) ----
        float alpha[8];
#pragma unroll
        for (int i = 0; i < 8; ++i) {
            float v = valid ? sc[i] * scale : NEGINF;
            float mx = v;
#pragma unroll
            for (int off = 1; off < 16; off <<= 1)
                mx = fmaxf(mx, __shfl_xor(mx, off));
            const float mnew = fmaxf(m[i], mx);
            alpha[i] = __expf(m[i] - mnew);      // exp(-inf - finite) = 0
            const float p = __expf(v - mnew);
            float sum = p;
#pragma unroll
            for (int off = 1; off < 16; off <<= 1)
                sum += __shfl_xor(sum, off);
            l[i] = l[i] * alpha[i] + sum;
            m[i] = mnew;
            sc[i] = p;
        }
#pragma unroll
        for (int nt = 0; nt < 8; ++nt)
#pragma unroll
            for (int i = 0; i < 8; ++i) acc[nt][i] *= alpha[i];

        // ---- transpose P (C layout) -> A layout through LDS ----
#pragma unroll
        for (int i = 0; i < 8; ++i)
            plds[(i + half * 8) * 16 + lid] = sc[i];
        __syncthreads();
        v2f ap[4];
#pragma unroll
        for (int c = 0; c < 4; ++c)
            ap[c] = *(const v2f*)(plds + lid * 16 + 4 * c + 2 * half);
        __syncthreads();

        // ---- P(16x16) x V(16x128): 32 WMMAs ----
#pragma unroll
        for (int c = 0; c < 4; ++c) {
            const int k0 = 4 * c + 2 * half;
            const float* vr0 = (s0 + k0 == last)
                ? vnrow : vc + ((flatbase + k0) * HKVN + kv) * (size_t)DD;
            const float* vr1 = (s0 + k0 + 1 == last)
                ? vnrow : vc + ((flatbase + k0 + 1) * HKVN + kv) * (size_t)DD;
#pragma unroll
            for (int nt = 0; nt < 8; ++nt) {
                v2f bv;
                bv.x = vr0[nt * 16 + lid];
                bv.y = vr1[nt * 16 + lid];
                acc[nt] = wmma_f32(ap[c], bv, acc[nt]);
            }
        }
    }

    // ---- write partial: acc[16][128], m[16], l[16] ----
#pragma unroll
    for (int nt = 0; nt < 8; ++nt)
#pragma unroll
        for (int i = 0; i < 8; ++i)
            pb[(i + half * 8) * 128 + nt * 16 + lid] = acc[nt][i];
    if (lid == 0) {
#pragma unroll
        for (int i = 0; i < 8; ++i) {
            pb[16 * 128 + (i + half * 8)] = m[i];
            pb[16 * 128 + 16 + (i + half * 8)] = l[i];
        }
    }
}

// ---------------------------------------------------------------------------
// Combine the SPLITS partials per (b, kv), emit attn output [B, H*D]
// ---------------------------------------------------------------------------
__global__ void __launch_bounds__(256)
combine_kernel(const float* __restrict__ part, float* __restrict__ attn) {
    const int b = blockIdx.x, kv = blockIdx.y;
    const int t = threadIdx.x;
    const float NEGINF = -__builtin_inff();
    __shared__ float Msh[16], Lsh[16];
    const float* pb0 = part + (size_t)(b * HKVN + kv) * SPLITS * PART_STRIDE;
    if (t < 16) {
        float M = NEGINF;
        for (int s = 0; s < SPLITS; ++s)
            M = fmaxf(M, pb0[s * PART_STRIDE + 2048 + t]);
        float L = 0.f;
        for (int s = 0; s < SPLITS; ++s) {
            const float ms = pb0[s * PART_STRIDE + 2048 + t];
            if (ms > NEGINF)
                L += pb0[s * PART_STRIDE + 2064 + t] * __expf(ms - M);
        }
        Msh[t] = M;
        Lsh[t] = (L > 0.f) ? L : 1.f;
    }
    __syncthreads();
    for (int e = t; e < 16 * 128; e += 256) {
        const int g = e >> 7, d = e & 127;
        float sum = 0.f;
        for (int s = 0; s < SPLITS; ++s) {
            const float ms = pb0[s * PART_STRIDE + 2048 + g];
            if (ms > NEGINF)
                sum += __expf(ms - Msh[g]) * pb0[s * PART_STRIDE + g * 128 + d];
        }
        attn[(size_t)b * (HH * DD) + (kv * 16 + g) * DD + d] = sum / Lsh[g];
    }
}

// ---------------------------------------------------------------------------
extern "C" void kernel_launch(void* const* d_in, const int* in_sizes, int n_in,
                              void* d_out, int out_size, void* d_ws, size_t ws_size,
                              hipStream_t stream) {
    const float* hs     = (const float*)d_in[0];   // [B, HID]
    const float* wqkv   = (const float*)d_in[1];   // [4608, HID]
    const float* wdense = (const float*)d_in[2];   // [HID, H*D]
    const float* kc     = (const float*)d_in[3];   // [NBLK, 16, HKV, D]
    const float* vc     = (const float*)d_in[4];
    const int*   bt     = (const int*)d_in[5];     // [B, 256]
    const int*   lens   = (const int*)d_in[7];     // [B]
    float* out = (float*)d_out;                    // [B, HID]

    float* qkv  = (float*)d_ws;                           // B*QKVN
    float* knew = qkv  + (size_t)BB * QKVN;               // B*HKV*D
    float* vnew = knew + (size_t)BB * HKVN * DD;          // B*HKV*D
    float* part = vnew + (size_t)BB * HKVN * DD;          // B*HKV*SPLITS*2080
    float* attn = part + (size_t)BB * HKVN * SPLITS * PART_STRIDE;  // B*H*D

    // 1) QKV projection
    gemm_wmma_kernel<<<dim3(BB / 16, QKVN / 16), dim3(32), 0, stream>>>(
        hs, wqkv, qkv, QKVN, HIDN);
    // 2) RoPE + export current-token k/v
    rope_kernel<<<dim3(BB), dim3(1024), 0, stream>>>(qkv, knew, vnew, lens);
    // 3) Paged flash-decode attention (split-K over the sequence)
    attn_kernel<<<dim3(BB, HKVN, SPLITS), dim3(32), 0, stream>>>(
        qkv, knew, vnew, kc, vc, bt, lens, part);
    // 4) Combine splits
    combine_kernel<<<dim3(BB, HKVN), dim3(256), 0, stream>>>(part, attn);
    // 5) Output projection
    gemm_wmma_kernel<<<dim3(BB / 16, HIDN / 16), dim3(32), 0, stream>>>(
        attn, wdense, out, HIDN, HIDN);
}